// VariationalGCNEncoder_84129819394640
// MI455X (gfx1250) — compile-verified
//
#include <hip/hip_runtime.h>
#include <hip/hip_bf16.h>

typedef __attribute__((ext_vector_type(2))) float v2f;
typedef __attribute__((ext_vector_type(8))) float v8f;

#define C_IN    256
#define C_HID   128
#define C_OUT   64
#define SCAN_T  1024

#define WMMA_F32(a, b, c) __builtin_amdgcn_wmma_f32_16x16x4_f32( \
    false, (a), false, (b), (short)0, (c), false, false)

// ---------------------------------------------------------------------------
// Degree histogram (integer) + rsqrt normalization
// ---------------------------------------------------------------------------
__global__ void zero_u32_kernel(unsigned* p, int n) {
    int i = blockIdx.x * blockDim.x + threadIdx.x;
    if (i < n) p[i] = 0u;
}

__global__ void deg_count_kernel(const int* __restrict__ edge_index, unsigned* cnt, int e) {
    int i = blockIdx.x * blockDim.x + threadIdx.x;
    if (i < e) atomicAdd(&cnt[edge_index[e + i]], 1u);   // dst = edge_index[1][i]
}

__global__ void dis_kernel(const unsigned* __restrict__ cnt, float* dis, int n) {
    int i = blockIdx.x * blockDim.x + threadIdx.x;
    if (i < n) dis[i] = rsqrtf((float)(cnt[i] + 1u));    // +1 self-loop
}

// ---------------------------------------------------------------------------
// Single-block exclusive prefix scan over per-node counts -> rowptr & cursor
// ---------------------------------------------------------------------------
__global__ __launch_bounds__(SCAN_T)
void scan_kernel(const unsigned* __restrict__ cnt, unsigned* __restrict__ rowptr,
                 unsigned* __restrict__ cursor, int n) {
    __shared__ unsigned sums[SCAN_T];
    int t = threadIdx.x;
    int per = (n + SCAN_T - 1) / SCAN_T;
    int lo = t * per;
    int hi = min(lo + per, n);
    unsigned s = 0;
    for (int i = lo; i < hi; ++i) s += cnt[i];
    sums[t] = s;
    __syncthreads();
    for (int off = 1; off < SCAN_T; off <<= 1) {
        unsigned v = (t >= off) ? sums[t - off] : 0u;
        __syncthreads();
        sums[t] += v;
        __syncthreads();
    }
    unsigned base = (t == 0) ? 0u : sums[t - 1];
    for (int i = lo; i < hi; ++i) {
        unsigned c = cnt[i];
        rowptr[i] = base;
        cursor[i] = base;
        base += c;
    }
    if (t == SCAN_T - 1) rowptr[n] = base;
}

__global__ void csr_fill_kernel(const int* __restrict__ edge_index,
                                unsigned* cursor, int* __restrict__ csr_src, int e) {
    int i = blockIdx.x * blockDim.x + threadIdx.x;
    if (i < e) {
        int src = edge_index[i];
        int dst = edge_index[e + i];
        unsigned pos = atomicAdd(&cursor[dst], 1u);
        csr_src[pos] = src;
    }
}

// ---------------------------------------------------------------------------
// FP32 WMMA GEMM, N-blocked x4, software-pipelined (distance 1).
// C[n, OUTC] = A[n, K] @ W[OUTC, K]^T.  One wave = 16x64 strip (4 tiles, 4
// independent accumulators).  launch_bounds(256,1) relaxes the VGPR budget so
// next-iteration operands live in distinct registers (no wait serialization).
// ---------------------------------------------------------------------------
template <int K>
__global__ __launch_bounds__(256, 1)
void gemm_wmma_x4_kernel(const float* __restrict__ A,
                         const float* __restrict__ W,
                         float* __restrict__ C,
                         int OUTC, int tnbCount, int wavesTotal) {
    int wave = (int)((blockIdx.x * blockDim.x + threadIdx.x) >> 5);
    int lane = threadIdx.x & 31;
    if (wave >= wavesTotal) return;           // wave-uniform: EXEC stays all-ones

    int tm  = wave / tnbCount;
    int tnb = wave - tm * tnbCount;           // 64-column block index
    int l   = lane & 15;
    int hi  = (lane >> 4) << 1;               // K sub-offset per half-wave: 0 or 2

    const float* a_ptr = A + (size_t)(tm * 16 + l) * K + hi;
    const float* b_ptr = W + (size_t)(tnb * 64 + l) * K + hi;
    const size_t bS = (size_t)16 * K;

    // pipeline prologue
    v2f a  = *(const v2f*)(a_ptr);
    v2f q0 = *(const v2f*)(b_ptr);
    v2f q1 = *(const v2f*)(b_ptr + bS);
    v2f q2 = *(const v2f*)(b_ptr + 2 * bS);
    v2f q3 = *(const v2f*)(b_ptr + 3 * bS);

    v8f c0 = {}, c1 = {}, c2 = {}, c3 = {};
#pragma unroll 8
    for (int k = 0; k < K; k += 4) {
        v2f an = a, n0 = q0, n1 = q1, n2 = q2, n3 = q3;
        if (k + 4 < K) {                      // prefetch next k-step
            an = *(const v2f*)(a_ptr + k + 4);
            n0 = *(const v2f*)(b_ptr + k + 4);
            n1 = *(const v2f*)(b_ptr + bS + k + 4);
            n2 = *(const v2f*)(b_ptr + 2 * bS + k + 4);
            n3 = *(const v2f*)(b_ptr + 3 * bS + k + 4);
        }
        c0 = WMMA_F32(a, q0, c0);
        c1 = WMMA_F32(a, q1, c1);
        c2 = WMMA_F32(a, q2, c2);
        c3 = WMMA_F32(a, q3, c3);
        a = an; q0 = n0; q1 = n1; q2 = n2; q3 = n3;
    }

    float* c_ptr = C + (size_t)(tm * 16 + (lane >> 4) * 8) * OUTC + tnb * 64 + l;
#pragma unroll
    for (int r = 0; r < 8; ++r) {
        c_ptr[(size_t)r * OUTC +  0] = c0[r];
        c_ptr[(size_t)r * OUTC + 16] = c1[r];
        c_ptr[(size_t)r * OUTC + 32] = c2[r];
        c_ptr[(size_t)r * OUTC + 48] = c3[r];
    }
}

// ---------------------------------------------------------------------------
// Layer-2 dual GEMM (Wmu & Wlv share the A operand): 8 accumulators,
// software-pipelined the same way.
// ---------------------------------------------------------------------------
__global__ __launch_bounds__(256, 1)
void gemm2_dual_kernel(const float* __restrict__ A,
                       const float* __restrict__ Wmu,
                       const float* __restrict__ Wlv,
                       float* __restrict__ HWmu,
                       float* __restrict__ HWlv,
                       int wavesTotal) {
    const int K = C_HID;                      // 128
    int wave = (int)((blockIdx.x * blockDim.x + threadIdx.x) >> 5);
    int lane = threadIdx.x & 31;
    if (wave >= wavesTotal) return;

    int tm = wave;
    int l  = lane & 15;
    int hi = (lane >> 4) << 1;

    const float* a_ptr = A + (size_t)(tm * 16 + l) * K + hi;
    const float* bm = Wmu + (size_t)l * K + hi;
    const float* bl = Wlv + (size_t)l * K + hi;
    const size_t bS = (size_t)16 * K;

    v2f a  = *(const v2f*)(a_ptr);
    v2f p0 = *(const v2f*)(bm);
    v2f p1 = *(const v2f*)(bm + bS);
    v2f p2 = *(const v2f*)(bm + 2 * bS);
    v2f p3 = *(const v2f*)(bm + 3 * bS);
    v2f q0 = *(const v2f*)(bl);
    v2f q1 = *(const v2f*)(bl + bS);
    v2f q2 = *(const v2f*)(bl + 2 * bS);
    v2f q3 = *(const v2f*)(bl + 3 * bS);

    v8f m0 = {}, m1 = {}, m2 = {}, m3 = {};
    v8f w0 = {}, w1 = {}, w2 = {}, w3 = {};
#pragma unroll 4
    for (int k = 0; k < K; k += 4) {
        v2f an = a, e0 = p0, e1 = p1, e2 = p2, e3 = p3;
        v2f f0 = q0, f1 = q1, f2 = q2, f3 = q3;
        if (k + 4 < K) {
            an = *(const v2f*)(a_ptr + k + 4);
            e0 = *(const v2f*)(bm + k + 4);
            e1 = *(const v2f*)(bm + bS + k + 4);
            e2 = *(const v2f*)(bm + 2 * bS + k + 4);
            e3 = *(const v2f*)(bm + 3 * bS + k + 4);
            f0 = *(const v2f*)(bl + k + 4);
            f1 = *(const v2f*)(bl + bS + k + 4);
            f2 = *(const v2f*)(bl + 2 * bS + k + 4);
            f3 = *(const v2f*)(bl + 3 * bS + k + 4);
        }
        m0 = WMMA_F32(a, p0, m0);
        m1 = WMMA_F32(a, p1, m1);
        m2 = WMMA_F32(a, p2, m2);
        m3 = WMMA_F32(a, p3, m3);
        w0 = WMMA_F32(a, q0, w0);
        w1 = WMMA_F32(a, q1, w1);
        w2 = WMMA_F32(a, q2, w2);
        w3 = WMMA_F32(a, q3, w3);
        a = an; p0 = e0; p1 = e1; p2 = e2; p3 = e3;
        q0 = f0; q1 = f1; q2 = f2; q3 = f3;
    }

    size_t rowBase = (size_t)(tm * 16 + (lane >> 4) * 8) * C_OUT + l;
    float* pm = HWmu + rowBase;
    float* pl = HWlv + rowBase;
#pragma unroll
    for (int r = 0; r < 8; ++r) {
        pm[(size_t)r * C_OUT +  0] = m0[r];
        pm[(size_t)r * C_OUT + 16] = m1[r];
        pm[(size_t)r * C_OUT + 32] = m2[r];
        pm[(size_t)r * C_OUT + 48] = m3[r];
        pl[(size_t)r * C_OUT +  0] = w0[r];
        pl[(size_t)r * C_OUT + 16] = w1[r];
        pl[(size_t)r * C_OUT + 32] = w2[r];
        pl[(size_t)r * C_OUT + 48] = w3[r];
    }
}

// ---------------------------------------------------------------------------
// Layer 1 aggregation (gather via CSR) + bias + ReLU, fused.
// ---------------------------------------------------------------------------
__global__ __launch_bounds__(256)
void agg1_kernel(const float* __restrict__ XW, const float* __restrict__ dis,
                 const int* __restrict__ csr_src, const unsigned* __restrict__ rowptr,
                 const float* __restrict__ b1, float* __restrict__ h, int n) {
    int i    = (int)((blockIdx.x * blockDim.x + threadIdx.x) >> 5);
    int lane = threadIdx.x & 31;
    if (i >= n) return;

    float di = dis[i];
    float s  = di * di;
    float4 acc = ((const float4*)XW)[(size_t)i * 32 + lane];
    acc.x *= s; acc.y *= s; acc.z *= s; acc.w *= s;

    unsigned j   = rowptr[i];
    unsigned end = rowptr[i + 1];
    for (; j + 2 <= end; j += 2) {
        int s0 = csr_src[j], s1 = csr_src[j + 1];
        float w0 = dis[s0] * di, w1 = dis[s1] * di;
        float4 u0 = ((const float4*)XW)[(size_t)s0 * 32 + lane];
        float4 u1 = ((const float4*)XW)[(size_t)s1 * 32 + lane];
        acc.x += u0.x * w0 + u1.x * w1;
        acc.y += u0.y * w0 + u1.y * w1;
        acc.z += u0.z * w0 + u1.z * w1;
        acc.w += u0.w * w0 + u1.w * w1;
    }
    if (j < end) {
        int s0 = csr_src[j];
        float w0 = dis[s0] * di;
        float4 u0 = ((const float4*)XW)[(size_t)s0 * 32 + lane];
        acc.x += u0.x * w0; acc.y += u0.y * w0;
        acc.z += u0.z * w0; acc.w += u0.w * w0;
    }

    float4 b = ((const float4*)b1)[lane];
    acc.x = fmaxf(acc.x + b.x, 0.0f);
    acc.y = fmaxf(acc.y + b.y, 0.0f);
    acc.z = fmaxf(acc.z + b.z, 0.0f);
    acc.w = fmaxf(acc.w + b.w, 0.0f);
    ((float4*)h)[(size_t)i * 32 + lane] = acc;
}

// ---------------------------------------------------------------------------
// Layer 2 aggregation, fused mu + logvar (half-wave each), bias, -> d_out.
// ---------------------------------------------------------------------------
__global__ __launch_bounds__(256)
void agg2_kernel(const float* __restrict__ HWmu, const float* __restrict__ HWlv,
                 const float* __restrict__ dis,
                 const int* __restrict__ csr_src, const unsigned* __restrict__ rowptr,
                 const float* __restrict__ bmu, const float* __restrict__ blv,
                 float* __restrict__ out, int n) {
    int i    = (int)((blockIdx.x * blockDim.x + threadIdx.x) >> 5);
    int lane = threadIdx.x & 31;
    if (i >= n) return;

    int c = lane & 15;
    bool is_mu = lane < 16;
    const float* HW = is_mu ? HWmu : HWlv;
    const float* bs = is_mu ? bmu : blv;

    float di = dis[i];
    float s  = di * di;
    float4 acc = ((const float4*)HW)[(size_t)i * 16 + c];
    acc.x *= s; acc.y *= s; acc.z *= s; acc.w *= s;

    unsigned j   = rowptr[i];
    unsigned end = rowptr[i + 1];
    for (; j + 2 <= end; j += 2) {
        int s0 = csr_src[j], s1 = csr_src[j + 1];
        float w0 = dis[s0] * di, w1 = dis[s1] * di;
        float4 u0 = ((const float4*)HW)[(size_t)s0 * 16 + c];
        float4 u1 = ((const float4*)HW)[(size_t)s1 * 16 + c];
        acc.x += u0.x * w0 + u1.x * w1;
        acc.y += u0.y * w0 + u1.y * w1;
        acc.z += u0.z * w0 + u1.z * w1;
        acc.w += u0.w * w0 + u1.w * w1;
    }
    if (j < end) {
        int s0 = csr_src[j];
        float w0 = dis[s0] * di;
        float4 u0 = ((const float4*)HW)[(size_t)s0 * 16 + c];
        acc.x += u0.x * w0; acc.y += u0.y * w0;
        acc.z += u0.z * w0; acc.w += u0.w * w0;
    }

    float4 b = ((const float4*)bs)[c];
    acc.x += b.x; acc.y += b.y; acc.z += b.z; acc.w += b.w;

    float4* dst = is_mu ? (float4*)out : (float4*)(out + (size_t)n * C_OUT);
    dst[(size_t)i * 16 + c] = acc;
}

// ---------------------------------------------------------------------------
// Launch
// ---------------------------------------------------------------------------
extern "C" void kernel_launch(void* const* d_in, const int* in_sizes, int n_in,
                              void* d_out, int out_size, void* d_ws, size_t ws_size,
                              hipStream_t stream) {
    const float* x   = (const float*)d_in[0];            // [N, 256]
    const int*   ei  = (const int*)d_in[1];              // [2, E]
    const float* W1  = (const float*)d_in[2];            // [128, 256]
    const float* b1  = (const float*)d_in[3];            // [128]
    const float* Wmu = (const float*)d_in[4];            // [64, 128]
    const float* bmu = (const float*)d_in[5];            // [64]
    const float* Wlv = (const float*)d_in[6];            // [64, 128]
    const float* blv = (const float*)d_in[7];            // [64]
    float* out = (float*)d_out;                          // [N*64 mu | N*64 lv]

    const int n = in_sizes[0] / C_IN;                    // 50000
    const int e = in_sizes[1] / 2;                       // 800000

    const size_t nA = (size_t)((n + 1 + 63) & ~63);
    const size_t eA = (size_t)((e + 63) & ~63);
    float*    dis     = (float*)d_ws;                    // n
    unsigned* cnt     = (unsigned*)(dis + nA);           // n
    unsigned* rowptr  = cnt + nA;                        // n+1
    unsigned* cursor  = rowptr + nA;                     // n
    int*      csr_src = (int*)(cursor + nA);             // e
    float*    XW      = (float*)(csr_src + eA);          // n*128
    float*    h       = XW + (size_t)n * C_HID;          // n*128
    float*    HWmu    = XW;                              // reuse after agg1
    float*    HWlv    = XW + (size_t)n * C_OUT;

    const int T = 256;

    // 1) CSR build + normalization (graph shared by all three aggregations)
    zero_u32_kernel<<<(n + T - 1) / T, T, 0, stream>>>(cnt, n);
    deg_count_kernel<<<(e + T - 1) / T, T, 0, stream>>>(ei, cnt, e);
    dis_kernel<<<(n + T - 1) / T, T, 0, stream>>>(cnt, dis, n);
    scan_kernel<<<1, SCAN_T, 0, stream>>>(cnt, rowptr, cursor, n);
    csr_fill_kernel<<<(e + T - 1) / T, T, 0, stream>>>(ei, cursor, csr_src, e);

    // 2) XW = x @ W1^T   (3125 tm x 2 column-blocks, one wave per 16x64 strip)
    {
        int tnb = C_HID / 64;                            // 2
        int wavesTotal = (n / 16) * tnb;                 // 6250
        gemm_wmma_x4_kernel<C_IN><<<(wavesTotal + 7) / 8, T, 0, stream>>>(
            x, W1, XW, C_HID, tnb, wavesTotal);
    }

    // 3) layer-1 gather-aggregation + bias + relu (one wave per node)
    agg1_kernel<<<(n * 32 + T - 1) / T, T, 0, stream>>>(XW, dis, csr_src, rowptr,
                                                        b1, h, n);

    // 4) fused dual GEMM: HWmu and HWlv share the A operand (one wave / 16 rows)
    {
        int wavesTotal = n / 16;                         // 3125
        gemm2_dual_kernel<<<(wavesTotal + 7) / 8, T, 0, stream>>>(
            h, Wmu, Wlv, HWmu, HWlv, wavesTotal);
    }

    // 5) layer-2 gather-aggregation + bias straight into d_out
    agg2_kernel<<<(n * 32 + T - 1) / T, T, 0, stream>>>(HWmu, HWlv, dis, csr_src,
                                                        rowptr, bmu, blv, out, n);
}